// Attention_17308718202977
// MI455X (gfx1250) — compile-verified
//
#include <hip/hip_runtime.h>
#include <hip/hip_bf16.h>
#include <math.h>

typedef __attribute__((ext_vector_type(2))) float v2f;
typedef __attribute__((ext_vector_type(8))) float v8f;

#define EMBED 768
#define HEADS 12
#define HEAD_DIM 64
#define BATCH 8
#define SEQ 1024
#define BS (BATCH * SEQ)          // 8192 rows
#define SCALE 0.125f              // 64^-0.5
#define SC_STRIDE 1028            // 1024 + 4 pad (16B skew per row -> distinct banks)

__device__ __forceinline__ v8f wmma_f32(v2f a, v2f b, v8f c) {
    // V_WMMA_F32_16X16X4_F32 : D(16x16,f32) = A(16x4,f32) * B(4x16,f32) + C
    return __builtin_amdgcn_wmma_f32_16x16x4_f32(
        /*neg_a=*/false, a, /*neg_b=*/false, b,
        /*c_mod=*/(short)0, c, /*reuse_a=*/false, /*reuse_b=*/false);
}

// ---------------------------------------------------------------------------
// Kernel 1: QKV projection.  X[BS,E] @ {q,k,v}_w[E,E] + bias  -> [B,H,S,D]
// grid = (BS/16, E/64), block = 384 (12 waves): wave = matrix*4 + colSubtile
// ---------------------------------------------------------------------------
__global__ void qkv_proj_kernel(const float* __restrict__ X,
                                const float* __restrict__ qw, const float* __restrict__ qb,
                                const float* __restrict__ kw, const float* __restrict__ kb,
                                const float* __restrict__ vw, const float* __restrict__ vb,
                                float* __restrict__ Q, float* __restrict__ K,
                                float* __restrict__ V) {
    const int wave = threadIdx.x >> 5;
    const int lane = threadIdx.x & 31;
    const int l15  = lane & 15;
    const int koff = (lane >> 4) << 1;          // 0 or 2
    const int m    = wave >> 2;                  // 0=q, 1=k, 2=v
    const int ct   = (blockIdx.y << 2) + (wave & 3);   // 0..47 (16-col tiles)
    const int rt   = blockIdx.x;                 // 0..511 (16-row tiles)
    const int row  = rt * 16 + l15;
    const int col  = ct * 16 + l15;

    const float* __restrict__ W  = (m == 0) ? qw : (m == 1) ? kw : vw;
    const float* __restrict__ Bv = (m == 0) ? qb : (m == 1) ? kb : vb;
    float* __restrict__ Out      = (m == 0) ? Q  : (m == 1) ? K  : V;

    v8f acc = {};
    const float* xrow = X + (size_t)row * EMBED;
    for (int k = 0; k < EMBED; k += 4) {
        v2f a = *(const v2f*)(xrow + k + koff);
        v2f b;
        b.x = W[(size_t)(k + koff) * EMBED + col];
        b.y = W[(size_t)(k + koff + 1) * EMBED + col];
        acc = wmma_f32(a, b, acc);
    }

    const int   mbase = (lane >> 4) << 3;        // 0 or 8
    const int   h     = col >> 6;                // head
    const int   d     = col & 63;
    const float bias  = Bv[col];
    for (int i = 0; i < 8; ++i) {
        int srow = rt * 16 + mbase + i;          // 0..BS-1
        int bidx = srow >> 10;                   // /SEQ
        int s    = srow & (SEQ - 1);
        Out[(((size_t)bidx * HEADS + h) * SEQ + s) * HEAD_DIM + d] = acc[i] + bias;
    }
}

// ---------------------------------------------------------------------------
// Kernel 2: attention. One block per (b,h, 16-query tile).
// block = 128 (4 waves). Dynamic LDS: scores[16][SC_STRIDE] + reductions.
// ---------------------------------------------------------------------------
__global__ void attn_kernel(const float* __restrict__ Q, const float* __restrict__ K,
                            const float* __restrict__ V,
                            float* __restrict__ attn, float* __restrict__ outp) {
    extern __shared__ float smem[];
    float (*sc)[SC_STRIDE] = (float(*)[SC_STRIDE])smem;            // 16 x 1028
    float* red     = smem + 16 * SC_STRIDE;                        // 16 x 8
    float* rowstat = red + 16 * 8;                                 // 16

    const int qt  = blockIdx.x & (SEQ / 16 - 1);   // query tile 0..63
    const int bh  = blockIdx.x >> 6;               // 0..B*H-1
    const int b   = bh / HEADS;
    const int h   = bh - b * HEADS;

    const int tid  = threadIdx.x;
    const int wave = tid >> 5;
    const int lane = tid & 31;
    const int l15  = lane & 15;
    const int koff = (lane >> 4) << 1;

    const float* __restrict__ Qb = Q + (size_t)bh * SEQ * HEAD_DIM;
    const float* __restrict__ Kb = K + (size_t)bh * SEQ * HEAD_DIM;
    const float* __restrict__ Vb = V + (size_t)bh * SEQ * HEAD_DIM;

    // ---- Phase 1: scores strip  S[16][1024] = scale * Qtile @ K^T ----
    const float* qrow = Qb + (size_t)(qt * 16 + l15) * HEAD_DIM;
    const int mbase = (lane >> 4) << 3;
    for (int t = 0; t < 16; ++t) {
        const int kt = wave * 16 + t;                 // key tile 0..63
        const float* krow = Kb + (size_t)(kt * 16 + l15) * HEAD_DIM;
        v8f acc = {};
        for (int d = 0; d < HEAD_DIM; d += 4) {
            v2f a  = *(const v2f*)(qrow + d + koff);  // A[m=l15][d..]
            v2f bb = *(const v2f*)(krow + d + koff);  // B[dk][n=l15] = K[n][dk]
            acc = wmma_f32(a, bb, acc);
        }
        for (int i = 0; i < 8; ++i)
            sc[mbase + i][kt * 16 + l15] = acc[i] * SCALE;
    }
    __syncthreads();

    // ---- Phase 2: row softmax over 16 x 1024 (8 threads per row) ----
    const int row = tid >> 3;        // 0..15
    const int l8  = tid & 7;         // 0..7
    float mx = -3.402823466e38f;
    for (int c = l8; c < SEQ; c += 8) mx = fmaxf(mx, sc[row][c]);
    red[row * 8 + l8] = mx;
    __syncthreads();
    if (l8 == 0) {
        float m2 = red[row * 8];
        for (int j = 1; j < 8; ++j) m2 = fmaxf(m2, red[row * 8 + j]);
        rowstat[row] = m2;
    }
    __syncthreads();
    const float rmax = rowstat[row];
    float sum = 0.f;
    for (int c = l8; c < SEQ; c += 8) {
        float e = __expf(sc[row][c] - rmax);
        sc[row][c] = e;
        sum += e;
    }
    red[row * 8 + l8] = sum;
    __syncthreads();
    if (l8 == 0) {
        float t = 0.f;
        for (int j = 0; j < 8; ++j) t += red[row * 8 + j];
        rowstat[row] = 1.0f / t;
    }
    __syncthreads();
    const float inv = rowstat[row];
    float* attnRow = attn + ((size_t)bh * SEQ + (size_t)(qt * 16 + row)) * SEQ;
    for (int c = l8; c < SEQ; c += 8) {
        float v = sc[row][c] * inv;
        sc[row][c] = v;
        attnRow[c] = v;                       // stream normalized attn to HBM
    }
    __syncthreads();

    // ---- Phase 3: O[16][64] = attn_strip @ V; wave owns 16-d-col tile ----
    const int dt = wave;                       // 0..3
    v8f acc = {};
    for (int k = 0; k < SEQ; k += 4) {
        v2f a;
        a.x = sc[l15][k + koff];
        a.y = sc[l15][k + koff + 1];
        v2f bb;
        bb.x = Vb[(size_t)(k + koff)     * HEAD_DIM + dt * 16 + l15];
        bb.y = Vb[(size_t)(k + koff + 1) * HEAD_DIM + dt * 16 + l15];
        acc = wmma_f32(a, bb, acc);
    }
    for (int i = 0; i < 8; ++i) {
        int q = qt * 16 + mbase + i;
        outp[((size_t)b * SEQ + q) * EMBED + h * HEAD_DIM + dt * 16 + l15] = acc[i];
    }
}

// ---------------------------------------------------------------------------
// Kernel 3: output projection. outp[BS,E] @ o_w[E,E] + o_b -> out[BS,E]
// grid = (BS/16, E/64), block = 128 (4 waves, one 16-col tile each)
// ---------------------------------------------------------------------------
__global__ void o_proj_kernel(const float* __restrict__ Xp,
                              const float* __restrict__ ow, const float* __restrict__ ob,
                              float* __restrict__ Out) {
    const int wave = threadIdx.x >> 5;
    const int lane = threadIdx.x & 31;
    const int l15  = lane & 15;
    const int koff = (lane >> 4) << 1;
    const int ct   = (blockIdx.y << 2) + wave;  // 0..47
    const int rt   = blockIdx.x;
    const int row  = rt * 16 + l15;
    const int col  = ct * 16 + l15;

    v8f acc = {};
    const float* xrow = Xp + (size_t)row * EMBED;
    for (int k = 0; k < EMBED; k += 4) {
        v2f a = *(const v2f*)(xrow + k + koff);
        v2f b;
        b.x = ow[(size_t)(k + koff) * EMBED + col];
        b.y = ow[(size_t)(k + koff + 1) * EMBED + col];
        acc = wmma_f32(a, b, acc);
    }
    const int   mbase = (lane >> 4) << 3;
    const float bias  = ob[col];
    for (int i = 0; i < 8; ++i) {
        int r = rt * 16 + mbase + i;
        Out[(size_t)r * EMBED + col] = acc[i] + bias;
    }
}

extern "C" void kernel_launch(void* const* d_in, const int* in_sizes, int n_in,
                              void* d_out, int out_size, void* d_ws, size_t ws_size,
                              hipStream_t stream) {
    (void)in_sizes; (void)n_in; (void)out_size; (void)ws_size;
    const float* X  = (const float*)d_in[0];
    const float* qw = (const float*)d_in[1];
    const float* qb = (const float*)d_in[2];
    const float* kw = (const float*)d_in[3];
    const float* kb = (const float*)d_in[4];
    const float* vw = (const float*)d_in[5];
    const float* vb = (const float*)d_in[6];
    const float* ow = (const float*)d_in[7];
    const float* ob = (const float*)d_in[8];

    const size_t perMat = (size_t)BATCH * HEADS * SEQ * HEAD_DIM;   // 6,291,456 f32
    float* Q    = (float*)d_ws;
    float* K    = Q + perMat;
    float* V    = K + perMat;
    float* outp = V + perMat;                                       // [B,S,E]

    float* out  = (float*)d_out;                                    // [B,S,E]
    float* attn = out + (size_t)BS * EMBED;                         // [B,H,S,S]

    // 1) QKV projection
    {
        dim3 grid(BS / 16, EMBED / 64);
        qkv_proj_kernel<<<grid, 384, 0, stream>>>(X, qw, qb, kw, kb, vw, vb, Q, K, V);
    }
    // 2) attention (scores + softmax + attn@V), dynamic LDS ~66 KB
    {
        const size_t smem = (size_t)(16 * SC_STRIDE + 16 * 8 + 16) * sizeof(float);
        dim3 grid(BATCH * HEADS * (SEQ / 16));
        attn_kernel<<<grid, 128, smem, stream>>>(Q, K, V, attn, outp);
    }
    // 3) output projection
    {
        dim3 grid(BS / 16, EMBED / 64);
        o_proj_kernel<<<grid, 128, 0, stream>>>(outp, ow, ob, out);
    }
}